// mads_25116968747009
// MI455X (gfx1250) — compile-verified
//
#include <hip/hip_runtime.h>

typedef __attribute__((ext_vector_type(2))) float v2f;
typedef __attribute__((ext_vector_type(8))) float v8f;

namespace {
constexpr int kBatch  = 8;
constexpr int kN      = 4096;
constexpr int kDK     = 32;
constexpr int kDV     = 32;
constexpr int kKStage = 128;   // keys staged in LDS per iteration (8 subtiles of 16)
constexpr int kQTile  = 128;   // queries per workgroup (8 waves x 16)
constexpr int kWaves  = 8;
constexpr int kStages = kN / kKStage;  // 32
constexpr float kLog2e = 1.4426950408889634f;
}

// Gaussian-kernel attention (mads 'nwd'/'gauss' forward).
// Score (in base 2): e2 = log2e*(ks.qs - 0.5|ks|^2 - 0.5|qs|^2), computed
// entirely on the WMMA pipe: 8 K=4 chunks for the cross term (query operands
// pre-scaled by log2e) plus ONE extra K=4 chunk carrying the rank-2 bias
// (A=[k2,1,0,0], B=[-0.5*log2e, -0.5*q2*log2e,0,0]). Elementwise softmax is
// then p = exp2(med3(e2, 0, -inf)) -- two VALU ops per element.
// VALUE tiles are double-buffered via GLOBAL_LOAD_ASYNC_TO_LDS_B128
// (ASYNCcnt); KEY tiles are software-pipelined through VGPRs.
__global__ __launch_bounds__(256)
void mads_gauss_attn(const float* __restrict__ KEY,
                     const float* __restrict__ VALUE,
                     const float* __restrict__ QUERY,
                     const float* __restrict__ W,
                     float* __restrict__ OUT) {
  // sKey stride 34: even (8B-aligned float2 A-reads), 34*r mod 64 distinct.
  // sVal stride 36: row pitch 144B -> every async B128 destination 16B-aligned.
  __shared__ __align__(16) float sKey[kKStage][34];
  __shared__ __align__(16) float sVal[2][kKStage][36];  // double buffer (async)
  __shared__ float sK2p[2 * kKStage];      // per-half-row |ks|^2 partials
  __shared__ float sW[kDK];
  __shared__ float sP[kWaves][16][17];     // per-wave P tile for layout transpose

  const int t    = threadIdx.x;
  const int w    = t >> 5;        // wave id 0..7
  const int l    = t & 31;        // lane
  const int lr   = l & 15;        // lane % 16
  const int half = l >> 4;        // 0: lanes 0-15, 1: lanes 16-31
  const int off  = half << 1;     // K offset within a K=4 chunk (A/B layout)
  const int b     = blockIdx.y;
  const int qbase = blockIdx.x * kQTile;

  if (t < kDK) sW[t] = W[t];
  __syncthreads();

  // ---- Per-wave query tile: B-layout operands (log2e * qs^T) + bias chunk ----
  // B layout (4x16): VGPR0 holds K=4c (lanes 0-15) / K=4c+2 (lanes 16-31),
  // VGPR1 holds K+1; N = lane%16 in both halves.
  const int jq = qbase + w * 16 + lr;
  const float* qr = QUERY + ((size_t)b * kN + jq) * kDK;
  v2f qb[kDK / 4];
  float q2 = 0.f;
#pragma unroll
  for (int d = 0; d < kDK; d += 2) {
    float x0 = qr[d]     * sW[d];
    float x1 = qr[d + 1] * sW[d + 1];
    q2 = fmaf(x0, x0, q2);              // |qs|^2 (unscaled)
    q2 = fmaf(x1, x1, q2);
    if (((d >> 1) & 1) == half) {       // d % 4 == off -> this lane's K slots
      qb[d >> 2] = (v2f){x0 * kLog2e, x1 * kLog2e};
    }
  }
  // Bias B chunk: K=0 -> -0.5*log2e, K=1 -> -0.5*log2e*q2_j, K=2,3 -> 0.
  v2f qbx;
  qbx.x = half ? 0.f : (-0.5f * kLog2e);
  qbx.y = half ? 0.f : (-0.5f * kLog2e * q2);

  // ---- staging helpers ----
  // VALUE: 128 rows x 8 float4 chunks = 1024 b128 chunks, 4 per thread,
  // copied straight into LDS with async loads (no VGPR round trip).
  auto issue_val_async = [&](int kb2, int buf) {
    const float* srcbase = VALUE + ((size_t)b * kN + kb2) * kDV;
#pragma unroll
    for (int rep = 0; rep < 4; ++rep) {
      const int idx = rep * 256 + t;
      const int r   = idx >> 3;        // value row within stage
      const int cg  = idx & 7;         // float4 chunk within row
      const uint32_t lds = (uint32_t)(uintptr_t)(&sVal[buf][r][cg * 4]);
      const uint64_t ga  = (uint64_t)(uintptr_t)(srcbase + r * kDV + cg * 4);
      asm volatile("global_load_async_to_lds_b128 %0, %1, off"
                   :: "v"(lds), "v"(ga) : "memory");
    }
  };

  // KEY: each thread owns half a row (16 floats); loads pipelined in VGPRs.
  const int krow  = t >> 1;
  const int kcol0 = (t & 1) * 16;
  float4 kreg[4];
  auto load_key = [&](int kb2) {
    const float4* p = reinterpret_cast<const float4*>(
        KEY + ((size_t)b * kN + kb2 + krow) * kDK + kcol0);
#pragma unroll
    for (int u = 0; u < 4; ++u) kreg[u] = p[u];
  };
  auto publish_key = [&]() {
    float s2 = 0.f;
#pragma unroll
    for (int u = 0; u < 4; ++u) {
      const int c = kcol0 + 4 * u;
      float y0 = kreg[u].x * sW[c + 0];
      float y1 = kreg[u].y * sW[c + 1];
      float y2 = kreg[u].z * sW[c + 2];
      float y3 = kreg[u].w * sW[c + 3];
      sKey[krow][c + 0] = y0;
      sKey[krow][c + 1] = y1;
      sKey[krow][c + 2] = y2;
      sKey[krow][c + 3] = y3;
      s2 = fmaf(y0, y0, s2); s2 = fmaf(y1, y1, s2);
      s2 = fmaf(y2, y2, s2); s2 = fmaf(y3, y3, s2);
    }
    sK2p[t] = s2;   // partial for row (t>>1); pair-summed at use
  };

  v8f accLo = {0.f, 0.f, 0.f, 0.f, 0.f, 0.f, 0.f, 0.f};  // out[j, 0:16]
  v8f accHi = {0.f, 0.f, 0.f, 0.f, 0.f, 0.f, 0.f, 0.f};  // out[j, 16:32]
  float lsum = 0.f;                                       // half-wave partial of sum_i p

  // ---- prologue: start stage 0 ----
  load_key(0);
  issue_val_async(0, 0);

  for (int s = 0; s < kStages; ++s) {
    const int cur = s & 1;
    publish_key();                       // sKey/sK2p for stage s (buf reuse safe
                                         // past previous iteration's end barrier)
    if (s + 1 < kStages) {
      load_key((s + 1) * kKStage);                 // overlap with compute
      issue_val_async((s + 1) * kKStage, cur ^ 1); // overlap with compute
      // ASYNC ops retire in order per wave: <=4 outstanding means stage-s
      // VALUE chunks have landed while stage-(s+1) remains in flight.
      asm volatile("s_wait_asynccnt 0x4" ::: "memory");
    } else {
      asm volatile("s_wait_asynccnt 0x0" ::: "memory");
    }
    __syncthreads();                     // stage s (sKey, sK2p, sVal[cur]) live

    for (int it = 0; it < kKStage / 16; ++it) {
      const int ib = it * 16;

      // Bias A chunk for this key subtile: cols K=0 -> k2_i, K=1 -> 1, else 0.
      const float k2row = sK2p[2 * (ib + lr)] + sK2p[2 * (ib + lr) + 1];
      v2f kax;
      kax.x = half ? 0.f : k2row;
      kax.y = half ? 0.f : 1.0f;

      // ---- GEMM1: e2 = log2e*(ks.qs - 0.5|ks|^2 - 0.5|qs|^2), 9 WMMAs ----
      v8f S = {0.f, 0.f, 0.f, 0.f, 0.f, 0.f, 0.f, 0.f};
#pragma unroll
      for (int c = 0; c < kDK / 4; ++c) {
        // A layout (16x4): M = lane%16, VGPR0 = K=4c+off, VGPR1 = K+1
        v2f ka = *reinterpret_cast<const v2f*>(&sKey[ib + lr][4 * c + off]);
        S = __builtin_amdgcn_wmma_f32_16x16x4_f32(
                false, ka, false, qb[c], (short)0, S, false, false);
      }
      S = __builtin_amdgcn_wmma_f32_16x16x4_f32(
              false, kax, false, qbx, (short)0, S, false, false);

      // ---- p = exp2(min(e2,0)) via med3; accumulate l; P->LDS ----
      // D layout: S[r] holds (i = r + 8*half, j = lane%16)
      float psum = 0.f;
#pragma unroll
      for (int r = 0; r < 8; ++r) {
        const float e2 = __builtin_amdgcn_fmed3f(S[r], 0.f,
                                                 -__builtin_huge_valf());
        const float p = __builtin_amdgcn_exp2f(e2);
        psum += p;
        sP[w][r + 8 * half][lr] = p;
      }
      lsum += psum;

      // Same-wave LDS RAW: DS ops are in-order per wave; wait for data return.
      __builtin_amdgcn_wave_barrier();
      asm volatile("s_wait_dscnt 0" ::: "memory");

      // ---- GEMM2: acc(16 queries x 16 dv) += P^T x V_tile, K=16 ----
#pragma unroll
      for (int c = 0; c < 4; ++c) {
        const int k0 = 4 * c + off;
        // A layout: A[j=lane%16, K] = P[i=k0..k0+1, j]
        v2f pa  = (v2f){sP[w][k0][lr], sP[w][k0 + 1][lr]};
        // B layout: B[K, dv=lane%16] = V[ib+k0..+1, dv]
        v2f vLo = (v2f){sVal[cur][ib + k0][lr],      sVal[cur][ib + k0 + 1][lr]};
        v2f vHi = (v2f){sVal[cur][ib + k0][16 + lr], sVal[cur][ib + k0 + 1][16 + lr]};
        accLo = __builtin_amdgcn_wmma_f32_16x16x4_f32(
                    false, pa, false, vLo, (short)0, accLo, false, false);
        accHi = __builtin_amdgcn_wmma_f32_16x16x4_f32(
                    false, pa, false, vHi, (short)0, accHi, false, false);
      }
    }
    __syncthreads();   // all waves done with stage s LDS before next overwrite
  }

  // ---- Normalize: combine half-wave partial sums, divide, store ----
  float ltot = lsum + __shfl_xor(lsum, 16, 32);   // full sum for j = lane%16
#pragma unroll
  for (int r = 0; r < 8; ++r) {
    const int jj  = r + 8 * half;                 // D layout row for acc[r]
    float dnm = __shfl(ltot, jj, 32);             // denom lives at lane jj
    float inv = 1.f / dnm;
    const int jrow = qbase + w * 16 + jj;
    float* o = OUT + ((size_t)b * kN + jrow) * kDV;
    o[lr]      = accLo[r] * inv;
    o[lr + 16] = accHi[r] * inv;
  }
}

extern "C" void kernel_launch(void* const* d_in, const int* in_sizes, int n_in,
                              void* d_out, int out_size, void* d_ws, size_t ws_size,
                              hipStream_t stream) {
  const float* KEY   = (const float*)d_in[0];
  const float* VALUE = (const float*)d_in[1];
  const float* QUERY = (const float*)d_in[2];
  const float* W     = (const float*)d_in[3];
  float* OUT = (float*)d_out;
  (void)in_sizes; (void)n_in; (void)out_size; (void)d_ws; (void)ws_size;

  dim3 grid(kN / kQTile, kBatch);   // (32, 8)
  dim3 block(256);                  // 8 wave32 waves
  mads_gauss_attn<<<grid, block, 0, stream>>>(KEY, VALUE, QUERY, W, OUT);
}